// TreeModel_81114752352408
// MI455X (gfx1250) — compile-verified
//
#include <hip/hip_runtime.h>

// ---------------------------------------------------------------------------
// TreeModel on gfx1250 (MI455X). After dead-code elimination only layer-3
// nodes 504..511 (x cols 4032..4095), layer-2 nodes 56..63, and all 8 layer-1
// nodes are live. BN3/BN2 reduce over the full batch -> kernel boundaries.
// All GEMMs use V_WMMA_F32_16X16X4_F32 (fp32 in/out, exact-precision path).
// ---------------------------------------------------------------------------

typedef float v2f __attribute__((ext_vector_type(2)));
typedef float v8f __attribute__((ext_vector_type(8)));

#define GATE_F 0.1f
#define BATCH_N 4096
#define EPS_F 1e-5f

static __device__ __forceinline__ v8f wmma4(v2f a, v2f b, v8f c) {
  // D = A(16x4,f32) * B(4x16,f32) + C(16x16,f32)
  return __builtin_amdgcn_wmma_f32_16x16x4_f32(
      /*neg_a=*/false, a, /*neg_b=*/false, b,
      /*c_mod=*/(short)0, c, /*reuse_a=*/false, /*reuse_b=*/false);
}

static __device__ __forceinline__ float gate_relu(float m, float l) {
  float r = m > 0.f ? m : 0.f;
  return GATE_F * r + (1.f - GATE_F) * l;
}

// ---------------------------------------------------------------------------
// k_init: zero the BN accumulators, precompute reduced (16x16) weights for
// layer-2 nodes 56..62 and layer-1 nodes 0..6 (tile(h,8) @ W == h @ sum-chunks)
// ---------------------------------------------------------------------------
__global__ void k_init(const float* __restrict__ W2m, const float* __restrict__ W2l,
                       const float* __restrict__ W1m, const float* __restrict__ W1l,
                       float* __restrict__ stats,
                       float* __restrict__ w2mr, float* __restrict__ w2lr,
                       float* __restrict__ w1mr, float* __restrict__ w1lr) {
  int tid = threadIdx.x;                  // 256 threads, 1 block
  if (tid < 64) stats[tid] = 0.f;         // sum3[16] sumsq3[16] sum2[16] sumsq2[16]
  for (int idx = tid; idx < 7 * 256; idx += 256) {
    int rn = idx >> 8;                    // reduced node 0..6
    int k  = (idx >> 4) & 15;             // K index 0..15
    int o  = idx & 15;                    // output feature
    float s2m = 0.f, s2l = 0.f, s1m = 0.f, s1l = 0.f;
    for (int c = 0; c < 8; ++c) {
      int i = 16 * c + k;
      s2m += W2m[(56 + rn) * 2048 + i * 16 + o];
      s2l += W2l[(56 + rn) * 2048 + i * 16 + o];
      s1m += W1m[rn * 2048 + i * 16 + o];
      s1l += W1l[rn * 2048 + i * 16 + o];
    }
    w2mr[idx] = s2m; w2lr[idx] = s2l; w1mr[idx] = s1m; w1lr[idx] = s1l;
  }
}

// ---------------------------------------------------------------------------
// k_layer3: o3 for nodes 504..511 only. Per wave: 16-row tile, per node a
// (16x8)@(8x16) GEMM = 2 WMMA K-steps, x2 for main/linear paths.
// Node 511 feeds the BN3 accumulators (sum / sumsq per feature).
// ---------------------------------------------------------------------------
__global__ void k_layer3(const float* __restrict__ x,
                         const float* __restrict__ W3m, const float* __restrict__ b3m,
                         const float* __restrict__ W3l, const float* __restrict__ b3l,
                         float* __restrict__ o3buf, float* __restrict__ stats) {
  const int lane = threadIdx.x & 31;
  const int wave = threadIdx.x >> 5;
  const int tile = blockIdx.x * (blockDim.x >> 5) + wave;
  const int r0 = tile * 16;
  const int lm = lane & 15, half = lane >> 4;

  for (int n = 0; n < 8; ++n) {
    const int node = 504 + n;
    const float* xrow = x + (size_t)(r0 + lm) * 4096 + 4032 + 8 * n;
    const float* wm = W3m + node * 128;   // (8,16) row-major
    const float* wl = W3l + node * 128;
    v8f accm = {}; v8f accl = {};
#pragma unroll
    for (int ks = 0; ks < 2; ++ks) {
      int kb = 4 * ks + 2 * half;
      v2f a;  a.x  = xrow[kb];            a.y  = xrow[kb + 1];
      v2f bm; bm.x = wm[kb * 16 + lm];    bm.y = wm[(kb + 1) * 16 + lm];
      v2f bl; bl.x = wl[kb * 16 + lm];    bl.y = wl[(kb + 1) * 16 + lm];
      accm = wmma4(a, bm, accm);
      accl = wmma4(a, bl, accl);
    }
    const float bmb = b3m[node * 16 + lm];
    const float blb = b3l[node * 16 + lm];
    float psum = 0.f, psq = 0.f;
#pragma unroll
    for (int v = 0; v < 8; ++v) {
      int row = r0 + v + 8 * half;
      float o = gate_relu(accm[v] + bmb, accl[v] + blb);
      o3buf[(size_t)row * 128 + n * 16 + lm] = o;
      psum += o; psq += o * o;
    }
    if (n == 7) {                         // node 511 -> BN3 stats
      atomicAdd(&stats[lm], psum);
      atomicAdd(&stats[16 + lm], psq);
    }
  }
}

// ---------------------------------------------------------------------------
// k_layer2: h3 = BN(o3[:,511]); nodes 56..62 via reduced 16x16 weights
// (4 WMMA K-steps each), node 63 via full (16x128)@(128x16) (32 K-steps).
// Node 63 feeds BN2 accumulators.
// ---------------------------------------------------------------------------
__global__ void k_layer2(const float* __restrict__ o3buf,
                         const float* __restrict__ g3, const float* __restrict__ be3,
                         const float* __restrict__ W2m, const float* __restrict__ b2m,
                         const float* __restrict__ W2l, const float* __restrict__ b2l,
                         const float* __restrict__ w2mr, const float* __restrict__ w2lr,
                         float* __restrict__ o2buf, float* __restrict__ stats) {
  __shared__ float sh[4][32];             // per wave: scale[16], shift[16]
  const int lane = threadIdx.x & 31;
  const int wave = threadIdx.x >> 5;
  const int tile = blockIdx.x * (blockDim.x >> 5) + wave;
  const int r0 = tile * 16;
  const int lm = lane & 15, half = lane >> 4;
  float* ss = sh[wave];

  if (lane < 16) {
    float mu  = stats[lane] * (1.f / BATCH_N);
    float var = stats[16 + lane] * (1.f / BATCH_N) - mu * mu;
    float sc  = rsqrtf(var + EPS_F) * g3[lane];
    ss[lane]      = sc;
    ss[16 + lane] = be3[lane] - mu * sc;
  }
  __syncthreads();

  const float* o3row = o3buf + (size_t)(r0 + lm) * 128;
  v2f ah[4];                              // h3 tile in A layout (K = feature)
#pragma unroll
  for (int ks = 0; ks < 4; ++ks) {
    int f = 4 * ks + 2 * half;
    ah[ks].x = o3row[112 + f]     * ss[f]     + ss[16 + f];
    ah[ks].y = o3row[112 + f + 1] * ss[f + 1] + ss[16 + f + 1];
  }

  // nodes 56..62: (16x16)@(16x16)
  for (int rn = 0; rn < 7; ++rn) {
    v8f accm = {}; v8f accl = {};
#pragma unroll
    for (int ks = 0; ks < 4; ++ks) {
      int kb = 4 * ks + 2 * half;
      v2f bm; bm.x = w2mr[rn * 256 + kb * 16 + lm]; bm.y = w2mr[rn * 256 + (kb + 1) * 16 + lm];
      v2f bl; bl.x = w2lr[rn * 256 + kb * 16 + lm]; bl.y = w2lr[rn * 256 + (kb + 1) * 16 + lm];
      accm = wmma4(ah[ks], bm, accm);
      accl = wmma4(ah[ks], bl, accl);
    }
    const float bmb = b2m[(56 + rn) * 16 + lm];
    const float blb = b2l[(56 + rn) * 16 + lm];
#pragma unroll
    for (int v = 0; v < 8; ++v) {
      int row = r0 + v + 8 * half;
      o2buf[(size_t)row * 128 + rn * 16 + lm] = gate_relu(accm[v] + bmb, accl[v] + blb);
    }
  }

  // node 63: (16x128)@(128x16), A = in2_last = o3 nodes 504..511 flat
  {
    const float* wm = W2m + 63 * 2048;
    const float* wl = W2l + 63 * 2048;
    v8f accm = {}; v8f accl = {};
    for (int ks = 0; ks < 32; ++ks) {
      int kb = 4 * ks + 2 * half;
      v2f a;  a.x  = o3row[kb];          a.y  = o3row[kb + 1];
      v2f bm; bm.x = wm[kb * 16 + lm];   bm.y = wm[(kb + 1) * 16 + lm];
      v2f bl; bl.x = wl[kb * 16 + lm];   bl.y = wl[(kb + 1) * 16 + lm];
      accm = wmma4(a, bm, accm);
      accl = wmma4(a, bl, accl);
    }
    const float bmb = b2m[63 * 16 + lm];
    const float blb = b2l[63 * 16 + lm];
    float psum = 0.f, psq = 0.f;
#pragma unroll
    for (int v = 0; v < 8; ++v) {
      int row = r0 + v + 8 * half;
      float o = gate_relu(accm[v] + bmb, accl[v] + blb);
      o2buf[(size_t)row * 128 + 7 * 16 + lm] = o;
      psum += o; psq += o * o;
    }
    atomicAdd(&stats[32 + lm], psum);
    atomicAdd(&stats[48 + lm], psq);
  }
}

// ---------------------------------------------------------------------------
// k_layer1_out: h2 = BN(o2 node 63); o1 nodes 0..6 (reduced 16x16 GEMMs),
// node 7 ((16x128)@(128x16)); then in0=o1 flat -> o0 (128->16 GEMM via LDS
// transpose bounce) -> out = o0 @ Wout + bout.
// ---------------------------------------------------------------------------
__global__ void k_layer1_out(const float* __restrict__ o2buf,
                             const float* __restrict__ g2, const float* __restrict__ be2,
                             const float* __restrict__ W1m, const float* __restrict__ b1m,
                             const float* __restrict__ W1l, const float* __restrict__ b1l,
                             const float* __restrict__ w1mr, const float* __restrict__ w1lr,
                             const float* __restrict__ W0m, const float* __restrict__ b0m,
                             const float* __restrict__ W0l, const float* __restrict__ b0l,
                             const float* __restrict__ Wout, const float* __restrict__ bout,
                             const float* __restrict__ stats, float* __restrict__ out) {
  // per-wave LDS: o1 tile (16x128), o0 tile (16x16), scale/shift (32)
  __shared__ float sh[4][16 * 128 + 16 * 16 + 32];
  const int lane = threadIdx.x & 31;
  const int wave = threadIdx.x >> 5;
  const int tile = blockIdx.x * (blockDim.x >> 5) + wave;
  const int r0 = tile * 16;
  const int lm = lane & 15, half = lane >> 4;
  float* lds_o1 = sh[wave];
  float* lds_o0 = sh[wave] + 16 * 128;
  float* ss     = sh[wave] + 16 * 128 + 16 * 16;

  if (lane < 16) {
    float mu  = stats[32 + lane] * (1.f / BATCH_N);
    float var = stats[48 + lane] * (1.f / BATCH_N) - mu * mu;
    float sc  = rsqrtf(var + EPS_F) * g2[lane];
    ss[lane]      = sc;
    ss[16 + lane] = be2[lane] - mu * sc;
  }
  __syncthreads();

  const float* o2row = o2buf + (size_t)(r0 + lm) * 128;
  v2f ah[4];                              // h2 tile in A layout
#pragma unroll
  for (int ks = 0; ks < 4; ++ks) {
    int f = 4 * ks + 2 * half;
    ah[ks].x = o2row[112 + f]     * ss[f]     + ss[16 + f];
    ah[ks].y = o2row[112 + f + 1] * ss[f + 1] + ss[16 + f + 1];
  }

  // o1 nodes 0..6
  for (int rn = 0; rn < 7; ++rn) {
    v8f accm = {}; v8f accl = {};
#pragma unroll
    for (int ks = 0; ks < 4; ++ks) {
      int kb = 4 * ks + 2 * half;
      v2f bm; bm.x = w1mr[rn * 256 + kb * 16 + lm]; bm.y = w1mr[rn * 256 + (kb + 1) * 16 + lm];
      v2f bl; bl.x = w1lr[rn * 256 + kb * 16 + lm]; bl.y = w1lr[rn * 256 + (kb + 1) * 16 + lm];
      accm = wmma4(ah[ks], bm, accm);
      accl = wmma4(ah[ks], bl, accl);
    }
    const float bmb = b1m[rn * 16 + lm];
    const float blb = b1l[rn * 16 + lm];
#pragma unroll
    for (int v = 0; v < 8; ++v)
      lds_o1[(v + 8 * half) * 128 + rn * 16 + lm] = gate_relu(accm[v] + bmb, accl[v] + blb);
  }

  // o1 node 7: (16x128)@(128x16), A = in1_last = o2 nodes 56..63 flat
  {
    const float* wm = W1m + 7 * 2048;
    const float* wl = W1l + 7 * 2048;
    v8f accm = {}; v8f accl = {};
    for (int ks = 0; ks < 32; ++ks) {
      int kb = 4 * ks + 2 * half;
      v2f a;  a.x  = o2row[kb];          a.y  = o2row[kb + 1];
      v2f bm; bm.x = wm[kb * 16 + lm];   bm.y = wm[(kb + 1) * 16 + lm];
      v2f bl; bl.x = wl[kb * 16 + lm];   bl.y = wl[(kb + 1) * 16 + lm];
      accm = wmma4(a, bm, accm);
      accl = wmma4(a, bl, accl);
    }
    const float bmb = b1m[7 * 16 + lm];
    const float blb = b1l[7 * 16 + lm];
#pragma unroll
    for (int v = 0; v < 8; ++v)
      lds_o1[(v + 8 * half) * 128 + 7 * 16 + lm] = gate_relu(accm[v] + bmb, accl[v] + blb);
  }
  __syncthreads();

  // o0 = gated(in0 @ W0{m,l}), in0 = o1 flat (K = 128), A read from LDS
  {
    v8f accm = {}; v8f accl = {};
    const float* a_row = lds_o1 + lm * 128;
    for (int ks = 0; ks < 32; ++ks) {
      int kb = 4 * ks + 2 * half;
      v2f a;  a.x  = a_row[kb];           a.y  = a_row[kb + 1];
      v2f bm; bm.x = W0m[kb * 16 + lm];   bm.y = W0m[(kb + 1) * 16 + lm];
      v2f bl; bl.x = W0l[kb * 16 + lm];   bl.y = W0l[(kb + 1) * 16 + lm];
      accm = wmma4(a, bm, accm);
      accl = wmma4(a, bl, accl);
    }
    const float bmb = b0m[lm];
    const float blb = b0l[lm];
#pragma unroll
    for (int v = 0; v < 8; ++v)
      lds_o0[(v + 8 * half) * 16 + lm] = gate_relu(accm[v] + bmb, accl[v] + blb);
  }
  __syncthreads();

  // out = o0 @ Wout (16x2) + bout : lane -> (row=lm, col=half)
  {
    float acc = bout[half];
#pragma unroll
    for (int o = 0; o < 16; ++o) acc += lds_o0[lm * 16 + o] * Wout[o * 2 + half];
    out[(size_t)(r0 + lm) * 2 + half] = acc;
  }
}

// ---------------------------------------------------------------------------
// Host launcher
// ---------------------------------------------------------------------------
extern "C" void kernel_launch(void* const* d_in, const int* in_sizes, int n_in,
                              void* d_out, int out_size, void* d_ws, size_t ws_size,
                              hipStream_t stream) {
  const float* x    = (const float*)d_in[0];
  const float* W3m  = (const float*)d_in[1];
  const float* b3m  = (const float*)d_in[2];
  const float* W3l  = (const float*)d_in[3];
  const float* b3l  = (const float*)d_in[4];
  const float* W2m  = (const float*)d_in[5];
  const float* b2m  = (const float*)d_in[6];
  const float* W2l  = (const float*)d_in[7];
  const float* b2l  = (const float*)d_in[8];
  const float* W1m  = (const float*)d_in[9];
  const float* b1m  = (const float*)d_in[10];
  const float* W1l  = (const float*)d_in[11];
  const float* b1l  = (const float*)d_in[12];
  const float* W0m  = (const float*)d_in[13];
  const float* b0m  = (const float*)d_in[14];
  const float* W0l  = (const float*)d_in[15];
  const float* b0l  = (const float*)d_in[16];
  const float* g3   = (const float*)d_in[17];
  const float* be3  = (const float*)d_in[18];
  const float* g2   = (const float*)d_in[19];
  const float* be2  = (const float*)d_in[20];
  const float* Wout = (const float*)d_in[21];
  const float* bout = (const float*)d_in[22];

  float* wsf    = (float*)d_ws;
  float* o3buf  = wsf;                      // 4096*128
  float* o2buf  = wsf + 524288;             // 4096*128
  float* stats  = wsf + 1048576;            // 64
  float* w2mr   = wsf + 1048640;            // 7*256 each
  float* w2lr   = w2mr + 1792;
  float* w1mr   = w2lr + 1792;
  float* w1lr   = w1mr + 1792;

  k_init<<<1, 256, 0, stream>>>(W2m, W2l, W1m, W1l, stats, w2mr, w2lr, w1mr, w1lr);
  k_layer3<<<64, 128, 0, stream>>>(x, W3m, b3m, W3l, b3l, o3buf, stats);
  k_layer2<<<64, 128, 0, stream>>>(o3buf, g3, be3, W2m, b2m, W2l, b2l, w2mr, w2lr,
                                   o2buf, stats);
  k_layer1_out<<<64, 128, 0, stream>>>(o2buf, g2, be2, W1m, b1m, W1l, b1l, w1mr, w1lr,
                                       W0m, b0m, W0l, b0l, Wout, bout, stats,
                                       (float*)d_out);
}